// MyTreeMultiRandom_40707700032016
// MI455X (gfx1250) — compile-verified
//
#include <hip/hip_runtime.h>
#include <hip/hip_bf16.h>

typedef float v2f __attribute__((ext_vector_type(2)));
typedef float v8f __attribute__((ext_vector_type(8)));

#define PIX   9216      // 96*96
#define IMGW  96
#define NCH   4
#define NTILE (PIX / 16) // 576
#define BIG2  1e20f      // sentinel for d^2 (sqrt -> 1e10 == reference BIG)

// One wave32 per block. Block b owns queries q = b*16 .. b*16+15.
// Lane l: query column qlocal = l & 15; candidate half = (l >= 16) ? rows 8..15 : rows 0..7
// of each 16x16 WMMA tile (matches the V_WMMA_F32_16X16X4_F32 C-matrix layout).
__global__ __launch_bounds__(32)
void knn_inpaint_wmma(const float* __restrict__ coded,
                      const int*   __restrict__ shutter,
                      float*       __restrict__ out)
{
    __shared__ int   shmask[PIX];     // 36 KB: full shutter map cached in LDS
    __shared__ float MD[32][12];      // per-lane top-3 d^2, 4 channels
    __shared__ int   MI[32][12];      // per-lane top-3 indices

    const int  lane   = threadIdx.x;        // 0..31
    const int  qlocal = lane & 15;
    const bool hiHalf = lane >= 16;
    const int  q      = blockIdx.x * 16 + qlocal;

    // ---- cache shutter map in LDS (vectorized, uniform trip count) ----
    {
        const int4* src = (const int4*)shutter;
        int4*       dst = (int4*)shmask;
        for (int i = lane; i < PIX / 4; i += 32) dst[i] = src[i];
    }
    __syncthreads();

    // ---- B fragment (query side), constant over the whole candidate sweep ----
    // B[K][q]: K0=-2*yq, K1=-2*xq (lanes 0-15), K2=1, K3=sq_q (lanes 16-31)
    const float yq  = ((float)(q / IMGW) / 96.0f - 0.5f) * 2.0f;
    const float xq  = ((float)(q % IMGW) / 96.0f - 0.5f) * 2.0f;
    const float sqq = yq * yq + xq * xq;
    v2f bfrag;
    bfrag.x = hiHalf ? 1.0f : (-2.0f * yq);
    bfrag.y = hiHalf ? sqq  : (-2.0f * xq);

    // ---- per-channel running top-3 (d^2 ascending; strict '<' keeps earliest
    //      index on ties, matching jax.lax.top_k tie-breaking) ----
    float t0[NCH], t1[NCH], t2[NCH];
    int   j0[NCH], j1[NCH], j2[NCH];
#pragma unroll
    for (int c = 0; c < NCH; ++c) {
        t0[c] = t1[c] = t2[c] = BIG2;
        j0[c] = j1[c] = j2[c] = 0;
    }

    // ---- sweep all candidate tiles: one K=4 f32 WMMA produces a 16x16 d^2 tile ----
#pragma unroll 2
    for (int tile = 0; tile < NTILE; ++tile) {
        // A fragment (candidate side): A[j] = (y_j, x_j, sq_j, 1) across K=0..3
        const int   jrow = tile * 16 + qlocal;          // this lane contributes row coords
        const float yj   = ((float)(jrow / IMGW) / 96.0f - 0.5f) * 2.0f;
        const float xj   = ((float)(jrow % IMGW) / 96.0f - 0.5f) * 2.0f;
        const float sqj  = yj * yj + xj * xj;
        v2f afrag;
        afrag.x = hiHalf ? sqj  : yj;
        afrag.y = hiHalf ? 1.0f : xj;

        v8f cacc = {};  // fresh accumulator: C == d^2 tile directly
        cacc = __builtin_amdgcn_wmma_f32_16x16x4_f32(
            /*neg_a=*/false, afrag, /*neg_b=*/false, bfrag,
            /*c_mod=*/(short)0, cacc, /*reuse_a=*/false, /*reuse_b=*/false);

        const int base = tile * 16 + (hiHalf ? 8 : 0);
#pragma unroll
        for (int v = 0; v < 8; ++v) {
            const float dv = cacc[v];      // d^2(candidate base+v, query q)
            const int   jj = base + v;
            const int   ch = shmask[jj];
#pragma unroll
            for (int c = 0; c < NCH; ++c) {
                const float dd = (ch == c) ? dv : BIG2;   // where(mask, d2, BIG^2)
                const bool b0 = dd < t0[c];
                const bool b1 = dd < t1[c];
                const bool b2 = dd < t2[c];
                const float n2 = b1 ? t1[c] : (b2 ? dd : t2[c]);
                const int   m2 = b1 ? j1[c] : (b2 ? jj : j2[c]);
                const float n1 = b0 ? t0[c] : (b1 ? dd : t1[c]);
                const int   m1 = b0 ? j0[c] : (b1 ? jj : j1[c]);
                const float n0 = b0 ? dd : t0[c];
                const int   m0 = b0 ? jj : j0[c];
                t2[c] = n2; j2[c] = m2;
                t1[c] = n1; j1[c] = m1;
                t0[c] = n0; j0[c] = m0;
            }
        }
    }

    // ---- publish per-lane lists; lanes 0-15 merge with partner lane+16 ----
#pragma unroll
    for (int c = 0; c < NCH; ++c) {
        MD[lane][c * 3 + 0] = t0[c]; MI[lane][c * 3 + 0] = j0[c];
        MD[lane][c * 3 + 1] = t1[c]; MI[lane][c * 3 + 1] = j1[c];
        MD[lane][c * 3 + 2] = t2[c]; MI[lane][c * 3 + 2] = j2[c];
    }
    __syncthreads();

    if (lane < 16) {
#pragma unroll
        for (int c = 0; c < NCH; ++c) {
            int pa = 0, pb = 0;
            float wsum = 0.0f, vsum = 0.0f;
#pragma unroll
            for (int k = 0; k < 3; ++k) {
                const float da = MD[lane][c * 3 + pa];
                const int   ia = MI[lane][c * 3 + pa];
                const float db = MD[lane + 16][c * 3 + pb];
                const int   ib = MI[lane + 16][c * 3 + pb];
                const bool  ta = (da < db) || ((da == db) && (ia < ib)); // index tie-break
                const float dk2 = ta ? da : db;
                const int   idx = ta ? ia : ib;
                pa += ta ? 1 : 0;
                pb += ta ? 0 : 1;

                const float dist = sqrtf(fmaxf(dk2, 0.0f));
                const float dk   = fmaxf(dist, 1e-12f);   // EPS clamp
                const float wk   = 1.0f / dk;             // P_POW == 1
                float val = coded[c * PIX + idx];
                val = (shmask[idx] == c) ? val : 0.0f;    // vals = coded * mask
                wsum += wk;
                vsum += val * wk;
            }
            const float interp = vsum / wsum;
            const float own    = coded[c * PIX + q];
            out[c * PIX + q]   = (shmask[q] == c) ? own : interp;
        }
    }
}

extern "C" void kernel_launch(void* const* d_in, const int* in_sizes, int n_in,
                              void* d_out, int out_size, void* d_ws, size_t ws_size,
                              hipStream_t stream) {
    const float* coded   = (const float*)d_in[0];   // (1,4,96,96) f32
    const int*   shutter = (const int*)d_in[1];     // (1,96,96)  i32
    float*       out     = (float*)d_out;           // (1,4,96,96) f32

    knn_inpaint_wmma<<<NTILE, 32, 0, stream>>>(coded, shutter, out);
}